// VectorQuantizerEMA_37598143709219
// MI455X (gfx1250) — compile-verified
//
#include <hip/hip_runtime.h>
#include <math.h>

typedef __attribute__((ext_vector_type(2))) float v2f;
typedef __attribute__((ext_vector_type(8))) float v8f;

#define N_CODES   1024
#define CODE_DIM  128
#define NPOS      32768          // 32 * 32 * 32
#define HW        1024           // 32*32 spatial per batch
#define SZ_T      4194304        // 32*128*32*32
#define SZ_I      32768
#define SZ_E      131072         // 1024*128
#define SZ_C      1024

// d_out layout (floats), reference tuple order
#define OFF_ZQST  0
#define OFF_LOSS  4194304
#define OFF_IDX   4194305
#define OFF_ZQ    4227073
#define OFF_ZE    8421377
#define OFF_NEMB  12615681
#define OFF_NCS   12746753
#define OFF_NEA   12747777

// workspace layout (floats)
#define WS_COUNTS 0              // 1024
#define WS_ESUM   1024           // 131072
#define WS_LOSS   132096         // 1
#define WS_ENORM  132100         // 1024
#define WS_IDX    133124         // 32768 ints
#define WS_NZERO  132097         // zero counts+esum+loss

#define DECAY 0.99f
#define ONE_M_DECAY 0.01f
#define EPSV 1e-5f
#define BETA 0.25f

// ---------------------------------------------------------------- zero ws
__global__ void vq_zero_kernel(float* p, int n) {
    int i = blockIdx.x * blockDim.x + threadIdx.x;
    if (i < n) p[i] = 0.0f;
}

// ---------------------------------------------------------------- ||e||^2
__global__ void vq_enorm_kernel(const float* __restrict__ embed,
                                float* __restrict__ enorm) {
    int n = blockIdx.x * blockDim.x + threadIdx.x;
    if (n < N_CODES) {
        const float* e = embed + (size_t)n * CODE_DIM;
        float s = 0.0f;
        #pragma unroll 8
        for (int k = 0; k < CODE_DIM; ++k) { float v = e[k]; s = fmaf(v, v, s); }
        enorm[n] = s;
    }
}

// ---------------------------------------------------------------- WMMA argmin
// grid: 256 blocks x 256 threads; each wave owns a 16-row tile of flat(z_e),
// iterates all 1024 codes. B staged via LDS (64-code chunks, padded stride).
#define CHUNK_N   64
#define LDS_STR   132   // 128 + 4 pad: rows 16B aligned, 4-bank rotation

__global__ __launch_bounds__(256) void vq_argmin_kernel(
    const float* __restrict__ zf, const float* __restrict__ embed,
    const float* __restrict__ enorm, int* __restrict__ idxi,
    float* __restrict__ idxf) {

    __shared__ float smem[CHUNK_N * LDS_STR];

    const int t    = threadIdx.x;
    const int lane = t & 31;
    const int wave = t >> 5;
    const int mrel = lane & 15;   // M (A) / N (B/C) index within tile
    const int half = lane >> 4;   // selects K pair

    const int m0 = blockIdx.x * 128 + wave * 16;   // 16-row tile, never crosses batch
    const int b  = m0 >> 10;
    const int s0 = m0 & 1023;

    // A fragment preload: A[m][k] = z_e[b][k][s], s = spatial index
    // lane layout: a[kk].x -> K = 4*kk + 2*half, a[kk].y -> K = 4*kk + 2*half + 1
    const float* zbase = zf + ((size_t)b << 17) + (size_t)(s0 + mrel);
    v2f a[32];
    #pragma unroll
    for (int kk = 0; kk < 32; ++kk) {
        const int k = kk * 4 + 2 * half;
        v2f av;
        av.x = zbase[(size_t)k << 10];
        av.y = zbase[(size_t)(k + 1) << 10];
        a[kk] = av;
    }

    float bestv[8];
    int   besti[8];
    #pragma unroll
    for (int v = 0; v < 8; ++v) { bestv[v] = 3.4e38f; besti[v] = 0; }

    for (int chunk = 0; chunk < N_CODES / CHUNK_N; ++chunk) {
        const int c0 = chunk * CHUNK_N;
        __syncthreads();
        {   // cooperative stage of 64 codes x 128 dims into padded LDS
            const int row = t >> 2;
            const int k0  = (t & 3) * 32;
            const float4* src = (const float4*)(embed + (size_t)(c0 + row) * CODE_DIM + k0);
            float4* dst = (float4*)(&smem[row * LDS_STR + k0]);
            #pragma unroll
            for (int q = 0; q < 8; ++q) dst[q] = src[q];
        }
        __syncthreads();

        for (int nt = 0; nt < CHUNK_N / 16; ++nt) {
            const float* brow = &smem[(nt * 16 + mrel) * LDS_STR + 2 * half];

            // prefetch all B fragments for this n-tile into registers so the
            // WMMA burst runs register-only (no per-WMMA s_wait_dscnt 0)
            v2f bf[32];
            #pragma unroll
            for (int kk = 0; kk < 32; ++kk)
                bf[kk] = *(const v2f*)(brow + kk * 4);

            // two independent accumulator chains to pipeline the XDL
            v8f acc0 = {};
            v8f acc1 = {};
            #pragma unroll
            for (int kk = 0; kk < 16; ++kk) {
                acc0 = __builtin_amdgcn_wmma_f32_16x16x4_f32(
                    false, a[2 * kk],     false, bf[2 * kk],     (short)0, acc0, false, false);
                acc1 = __builtin_amdgcn_wmma_f32_16x16x4_f32(
                    false, a[2 * kk + 1], false, bf[2 * kk + 1], (short)0, acc1, false, false);
            }

            const int n = c0 + nt * 16 + mrel;
            const float en = enorm[n];
            #pragma unroll
            for (int v = 0; v < 8; ++v) {
                const float d = fmaf(-2.0f, acc0[v] + acc1[v], en);
                if (d < bestv[v]) { bestv[v] = d; besti[v] = n; }
            }
        }
    }

    // reduce argmin across the 16 lanes sharing each row (halves are xor-closed)
    #pragma unroll
    for (int v = 0; v < 8; ++v) {
        float mv = bestv[v];
        int   mi = besti[v];
        #pragma unroll
        for (int mask = 1; mask < 16; mask <<= 1) {
            const float ov = __shfl_xor(mv, mask, 32);
            const int   oi = __shfl_xor(mi, mask, 32);
            if (ov < mv || (ov == mv && oi < mi)) { mv = ov; mi = oi; }
        }
        bestv[v] = mv; besti[v] = mi;
    }
    if (mrel == 0) {   // lanes 0 (rows 0..7) and 16 (rows 8..15)
        const int rowbase = m0 + half * 8;
        #pragma unroll
        for (int v = 0; v < 8; ++v) {
            idxi[rowbase + v] = besti[v];
            idxf[rowbase + v] = (float)besti[v];
        }
    }
}

// ---------------------------------------------------------------- scatter
// block = 256 threads = 2 positions x 128 channels
__global__ __launch_bounds__(256) void vq_scatter_kernel(
    const float* __restrict__ zf, const float* __restrict__ embed,
    const int* __restrict__ idxi, float* __restrict__ out,
    float* __restrict__ counts, float* __restrict__ esum,
    float* __restrict__ loss_sum) {

    __shared__ float red[256];
    const int t   = threadIdx.x;
    const int pos = blockIdx.x * 2 + (t >> 7);
    const int c   = t & 127;
    const int b   = pos >> 10;
    const int s   = pos & 1023;
    const int code = idxi[pos];

    const size_t zoff = ((size_t)(b * CODE_DIM + c) << 10) + (size_t)s;
    const float ze = zf[zoff];
    const float q  = embed[(size_t)code * CODE_DIM + c];

    out[OFF_ZQST + zoff] = q;   // value of straight-through output == z_q
    out[OFF_ZQ   + zoff] = q;
    out[OFF_ZE   + zoff] = ze;

    atomicAdd(&esum[(size_t)code * CODE_DIM + c], ze);
    if (c == 0) atomicAdd(&counts[code], 1.0f);

    const float d = q - ze;
    red[t] = d * d;
    __syncthreads();
    #pragma unroll
    for (int off = 128; off > 0; off >>= 1) {
        if (t < off) red[t] += red[t + off];
        __syncthreads();
    }
    if (t == 0) atomicAdd(loss_sum, red[0]);
}

// ---------------------------------------------------------------- finalize
__global__ __launch_bounds__(1024) void vq_finalize_kernel(
    const float* __restrict__ cluster_size, const float* __restrict__ embed_avg,
    const float* __restrict__ counts, const float* __restrict__ esum,
    const float* __restrict__ loss_sum, float* __restrict__ out) {

    __shared__ float red[1024];
    const int n = threadIdx.x;
    const float ncs = cluster_size[n] * DECAY + ONE_M_DECAY * counts[n];
    red[n] = ncs;
    __syncthreads();
    for (int off = 512; off > 0; off >>= 1) {
        if (n < off) red[n] += red[n + off];
        __syncthreads();
    }
    const float ntot  = red[0];
    const float denom = ntot + (float)N_CODES * EPSV;
    const float cs    = (ncs + EPSV) / denom * ntot;
    const float csafe = fmaxf(cs, EPSV);

    #pragma unroll 4
    for (int k = 0; k < CODE_DIM; ++k) {
        const size_t o = (size_t)n * CODE_DIM + k;
        const float nea = embed_avg[o] * DECAY + ONE_M_DECAY * esum[o];
        out[OFF_NEA + o] = nea;
        float en = nea / csafe;
        if (isnan(en))      en = 0.0f;
        else if (isinf(en)) en = (en > 0.0f) ? 1.0f : -1.0f;
        en = fminf(fmaxf(en, -2.0f), 2.0f);
        out[OFF_NEMB + o] = en;
    }
    out[OFF_NCS + n] = ncs;
    if (n == 0) out[OFF_LOSS] = BETA * loss_sum[0] / (float)SZ_T;
}

// ---------------------------------------------------------------- launch
extern "C" void kernel_launch(void* const* d_in, const int* in_sizes, int n_in,
                              void* d_out, int out_size, void* d_ws, size_t ws_size,
                              hipStream_t stream) {
    (void)in_sizes; (void)n_in; (void)out_size; (void)ws_size;
    const float* z_e          = (const float*)d_in[0];
    const float* embed        = (const float*)d_in[1];
    const float* cluster_size = (const float*)d_in[2];
    const float* embed_avg    = (const float*)d_in[3];
    float* out = (float*)d_out;
    float* ws  = (float*)d_ws;

    float* counts   = ws + WS_COUNTS;
    float* esum     = ws + WS_ESUM;
    float* loss_sum = ws + WS_LOSS;
    float* enorm    = ws + WS_ENORM;
    int*   idxi     = (int*)(ws + WS_IDX);

    vq_zero_kernel<<<(WS_NZERO + 255) / 256, 256, 0, stream>>>(ws, WS_NZERO);
    vq_enorm_kernel<<<(N_CODES + 255) / 256, 256, 0, stream>>>(embed, enorm);
    vq_argmin_kernel<<<NPOS / 128, 256, 0, stream>>>(z_e, embed, enorm, idxi, out + OFF_IDX);
    vq_scatter_kernel<<<NPOS / 2, 256, 0, stream>>>(z_e, embed, idxi, out, counts, esum, loss_sum);
    vq_finalize_kernel<<<1, 1024, 0, stream>>>(cluster_size, embed_avg, counts, esum, loss_sum, out);
}